// GraphNeuralNetwork_4114578670491
// MI455X (gfx1250) — compile-verified
//
#include <hip/hip_runtime.h>

typedef __attribute__((ext_vector_type(2))) float v2f;
typedef __attribute__((ext_vector_type(8))) float v8f;

// ---------- helpers ----------
__device__ __forceinline__ unsigned encOrd(float f) {
  unsigned u = __float_as_uint(f);
  return (u & 0x80000000u) ? ~u : (u | 0x80000000u);
}
__device__ __forceinline__ float decOrd(unsigned u) {
  return (u & 0x80000000u) ? __uint_as_float(u ^ 0x80000000u)
                           : __uint_as_float(~u);
}
__device__ __forceinline__ float lrelu(float e) { return e > 0.f ? e : 0.2f * e; }

// ---------- 1) GEMM: H = X(N x 128) @ W(128 x Nout), fp32 WMMA ----------
// Block = 128 threads (4 waves). Block tile: 16 (M) x 64 (N). K fixed at 128.
// W column panel (128x64, 32KB) and A tile (16x128, 8KB) staged in LDS.
__global__ __launch_bounds__(128)
void gat_gemm_wmma(const float* __restrict__ X, const float* __restrict__ W,
                   float* __restrict__ Hout, int N, int Nout) {
  __shared__ float sW[128 * 64];
  __shared__ float sA[16 * 128];
  const int tid  = threadIdx.x;
  const int wave = tid >> 5;
  const int lane = tid & 31;
  const int m0 = blockIdx.x * 16;
  const int nb = blockIdx.y * 64;

  // stage W[:, nb:nb+64] (128 rows x 64 cols) via float4, coalesced
  {
    const float4* Wv = reinterpret_cast<const float4*>(W);
    float4* sWv = reinterpret_cast<float4*>(sW);
    const int Nout4 = Nout >> 2;
    const int nb4 = nb >> 2;
#pragma unroll
    for (int i = 0; i < 16; ++i) {
      int idx4 = i * 128 + tid;        // 0..2047 float4 (16 per row)
      int row = idx4 >> 4;
      int c4  = idx4 & 15;
      sWv[idx4] = Wv[(size_t)row * Nout4 + nb4 + c4];
    }
  }
  // stage A rows [m0, m0+16) x 128 via float4
  {
    const float4* Xv = reinterpret_cast<const float4*>(X);
    float4* sAv = reinterpret_cast<float4*>(sA);
#pragma unroll
    for (int i = 0; i < 4; ++i) {
      int idx4 = i * 128 + tid;        // 0..511 float4 (32 per row)
      int row = idx4 >> 5;
      int c4  = idx4 & 31;
      int gr = m0 + row;
      if (gr >= N) gr = N - 1;         // safe clamp (rows never stored OOB)
      sAv[idx4] = Xv[(size_t)gr * 32 + c4];
    }
  }
  __syncthreads();

  // A frag layout (16x4 f32): lanes 0-15 -> M=lane, K={k0,k0+1}; lanes 16-31 -> K={k0+2,k0+3}
  // B frag layout (4x16 f32): lanes 0-15 -> N=lane, K={k0,k0+1}; lanes 16-31 -> K={k0+2,k0+3}
  const int mr   = lane & 15;
  const int koff = (lane >> 4) * 2;
  const int colL = wave * 16 + mr;     // local column in 0..63

  v8f acc = {0.f, 0.f, 0.f, 0.f, 0.f, 0.f, 0.f, 0.f};
#pragma unroll
  for (int k0 = 0; k0 < 128; k0 += 4) {
    v2f a, b;
    a[0] = sA[mr * 128 + k0 + koff];
    a[1] = sA[mr * 128 + k0 + koff + 1];
    b[0] = sW[(k0 + koff) * 64 + colL];
    b[1] = sW[(k0 + koff + 1) * 64 + colL];
    acc = __builtin_amdgcn_wmma_f32_16x16x4_f32(false, a, false, b,
                                                (short)0, acc, false, false);
  }

  // C/D layout: VGPR i -> (M=i, N=lane) for lanes 0-15, (M=i+8, N=lane-16) for 16-31
  const int col = nb + colL;
  const int rbase = m0 + (lane >> 4) * 8;
#pragma unroll
  for (int i = 0; i < 8; ++i) {
    int row = rbase + i;
    if (row < N) Hout[(size_t)row * Nout + col] = acc[i];
  }
}

// ---------- 2) per-node attention logits + self-loop max init ----------
__global__ __launch_bounds__(256)
void gat_node_pre(const float* __restrict__ Hf, const float* __restrict__ a_s,
                  const float* __restrict__ a_d, float* __restrict__ sl,
                  float* __restrict__ dl, unsigned* __restrict__ menc,
                  int N, int Hh, int C) {
  int t = blockIdx.x * blockDim.x + threadIdx.x;
  if (t >= N * Hh) return;
  int n = t / Hh, hh = t - n * Hh;
  const float* row = Hf + (size_t)n * (Hh * C) + hh * C;
  float s = 0.f, d = 0.f;
  for (int c = 0; c < C; ++c) {
    float v = row[c];
    s = __builtin_fmaf(v, a_s[hh * C + c], s);
    d = __builtin_fmaf(v, a_d[hh * C + c], d);
  }
  sl[t] = s;
  dl[t] = d;
  menc[t] = encOrd(lrelu(s + d));      // self-loop edge logit seeds the max
}

// ---------- 3) segment max over dst (ordered-uint float atomicMax) ----------
__global__ __launch_bounds__(256)
void gat_edge_max(const int* __restrict__ src, const int* __restrict__ dst,
                  const float* __restrict__ sl, const float* __restrict__ dl,
                  unsigned* __restrict__ menc, int E, int Hh) {
  int t = blockIdx.x * blockDim.x + threadIdx.x;
  if (t >= E) return;
  int s = src[t], d = dst[t];
  for (int hh = 0; hh < Hh; ++hh) {
    float e = lrelu(sl[s * Hh + hh] + dl[d * Hh + hh]);
    atomicMax(&menc[d * Hh + hh], encOrd(e));
  }
}

// ---------- 4) decode max, init denominator & accumulator with self loop ----------
__global__ __launch_bounds__(256)
void gat_node_mid(const float* __restrict__ Hf, const float* __restrict__ sl,
                  const float* __restrict__ dl, unsigned* __restrict__ menc,
                  float* __restrict__ den, float* __restrict__ acc,
                  int N, int Hh, int C) {
  int t = blockIdx.x * blockDim.x + threadIdx.x;
  if (t >= N * Hh) return;
  int n = t / Hh, hh = t - n * Hh;
  float m = decOrd(menc[t]);
  menc[t] = __float_as_uint(m);        // repurpose buffer as plain float max
  float e = lrelu(sl[t] + dl[t]);
  float ex = __expf(e - m);
  den[t] = ex;
  const size_t base = (size_t)n * (Hh * C) + hh * C;
  for (int c = 0; c < C; ++c) acc[base + c] = Hf[base + c] * ex;
}

// ---------- 5) wave-per-edge weighted aggregation ----------
// lanes stride channels; normalization by den deferred (den depends only on dst)
__global__ __launch_bounds__(256)
void gat_edge_agg(const int* __restrict__ src, const int* __restrict__ dst,
                  const float* __restrict__ sl, const float* __restrict__ dl,
                  const float* __restrict__ mf, float* __restrict__ den,
                  const float* __restrict__ Hf, float* __restrict__ acc,
                  int E, int Hh, int Cshift, int Nout) {
  int tid = blockIdx.x * blockDim.x + threadIdx.x;
  int edge = tid >> 5;
  int lane = tid & 31;
  if (edge >= E) return;
  int s = src[edge], d = dst[edge];
  const size_t sbase = (size_t)s * Nout;
  const size_t dbase = (size_t)d * Nout;
  const int nIter = Nout >> 5;
  const int Cmask = (1 << Cshift) - 1;
  for (int i = 0; i < nIter; ++i) {
    int c = lane + (i << 5);
    int hh = c >> Cshift;
    float e = lrelu(sl[s * Hh + hh] + dl[d * Hh + hh]);
    float ex = __expf(e - mf[d * Hh + hh]);
    atomicAdd(&acc[dbase + c], Hf[sbase + c] * ex);
    if ((c & Cmask) == 0) atomicAdd(&den[d * Hh + hh], ex);
  }
}

// ---------- 6a) normalize + bias + BatchNorm(eval) + ELU ----------
__global__ __launch_bounds__(256)
void gat_finalize_mid(const float* __restrict__ acc, const float* __restrict__ den,
                      const float* __restrict__ bias, const float* __restrict__ g,
                      const float* __restrict__ be, const float* __restrict__ rm,
                      const float* __restrict__ rv, float* __restrict__ out,
                      int N, int Nout, int Hh, int Cshift) {
  int t = blockIdx.x * blockDim.x + threadIdx.x;
  if (t >= N * Nout) return;
  int n = t / Nout, c = t - n * Nout;
  int hh = c >> Cshift;
  float v = acc[t] / den[n * Hh + hh] + bias[c];
  v = (v - rm[c]) * rsqrtf(rv[c] + 1e-5f) * g[c] + be[c];
  out[t] = v > 0.f ? v : __expf(v) - 1.f;   // ELU(alpha=1)
}

// ---------- 6b) last layer: normalize + bias (heads=1, mean == identity) ----------
__global__ __launch_bounds__(256)
void gat_finalize_last(const float* __restrict__ acc, const float* __restrict__ den,
                       const float* __restrict__ bias, float* __restrict__ out,
                       int N, int Nout) {
  int t = blockIdx.x * blockDim.x + threadIdx.x;
  if (t >= N * Nout) return;
  int n = t / Nout, c = t - n * Nout;
  out[t] = acc[t] / den[n] + bias[c];
}

// ---------- host-side layer driver ----------
static void run_gat_layer(const float* xin, const float* W, const float* a_s,
                          const float* a_d, float* hbuf, float* accbuf,
                          float* sl, float* dl, unsigned* menc, float* den,
                          const int* src, const int* dst, int N, int E,
                          int Nout, int Hh, int C, hipStream_t stream) {
  dim3 gg((N + 15) / 16, Nout / 64);
  gat_gemm_wmma<<<gg, 128, 0, stream>>>(xin, W, hbuf, N, Nout);
  int nh = N * Hh;
  gat_node_pre<<<(nh + 255) / 256, 256, 0, stream>>>(hbuf, a_s, a_d, sl, dl, menc, N, Hh, C);
  gat_edge_max<<<(E + 255) / 256, 256, 0, stream>>>(src, dst, sl, dl, menc, E, Hh);
  gat_node_mid<<<(nh + 255) / 256, 256, 0, stream>>>(hbuf, sl, dl, menc, den, accbuf, N, Hh, C);
  int Cshift = (C == 32) ? 5 : 6;
  long long tot = (long long)E * 32;
  gat_edge_agg<<<(unsigned)((tot + 255) / 256), 256, 0, stream>>>(
      src, dst, sl, dl, (const float*)menc, den, hbuf, accbuf, E, Hh, Cshift, Nout);
}

extern "C" void kernel_launch(void* const* d_in, const int* in_sizes, int n_in,
                              void* d_out, int out_size, void* d_ws, size_t ws_size,
                              hipStream_t stream) {
  const float* x   = (const float*)d_in[0];
  const int*   ei  = (const int*)d_in[1];
  const float* W1  = (const float*)d_in[2];
  const float* as1 = (const float*)d_in[3];
  const float* ad1 = (const float*)d_in[4];
  const float* b1  = (const float*)d_in[5];
  const float* g1  = (const float*)d_in[6];
  const float* be1 = (const float*)d_in[7];
  const float* rm1 = (const float*)d_in[8];
  const float* rv1 = (const float*)d_in[9];
  const float* W2  = (const float*)d_in[10];
  const float* as2 = (const float*)d_in[11];
  const float* ad2 = (const float*)d_in[12];
  const float* b2  = (const float*)d_in[13];
  const float* g2  = (const float*)d_in[14];
  const float* be2 = (const float*)d_in[15];
  const float* rm2 = (const float*)d_in[16];
  const float* rv2 = (const float*)d_in[17];
  const float* W3  = (const float*)d_in[18];
  const float* as3 = (const float*)d_in[19];
  const float* ad3 = (const float*)d_in[20];
  const float* b3  = (const float*)d_in[21];

  const int N = in_sizes[0] / 128;
  const int E = in_sizes[1] / 2;
  const int* src = ei;
  const int* dst = ei + E;

  // workspace layout (floats): 3 ping-pong feature buffers + node stats
  float* bufA = (float*)d_ws;                       // N*128
  float* bufB = bufA + (size_t)N * 128;             // N*128
  float* bufC = bufB + (size_t)N * 128;             // N*128 (aggregation acc)
  float* sl   = bufC + (size_t)N * 128;             // N*4
  float* dl   = sl + (size_t)N * 4;                 // N*4
  unsigned* menc = (unsigned*)(dl + (size_t)N * 4); // N*4 (max, enc->float)
  float* den  = (float*)(menc + (size_t)N * 4);     // N*4

  // Layer 1: GAT(128->32, H=4, concat) -> BN -> ELU ; output in bufA
  run_gat_layer(x, W1, as1, ad1, bufB, bufC, sl, dl, menc, den,
                src, dst, N, E, 128, 4, 32, stream);
  gat_finalize_mid<<<((N * 128) + 255) / 256, 256, 0, stream>>>(
      bufC, den, b1, g1, be1, rm1, rv1, bufA, N, 128, 4, 5);

  // Layer 2: GAT(128->32, H=4, concat) -> BN -> ELU ; output in bufB
  run_gat_layer(bufA, W2, as2, ad2, bufB, bufC, sl, dl, menc, den,
                src, dst, N, E, 128, 4, 32, stream);
  // note: finalize reads bufC only; bufB (h of layer 2) is dead, reuse as output
  gat_finalize_mid<<<((N * 128) + 255) / 256, 256, 0, stream>>>(
      bufC, den, b2, g2, be2, rm2, rv2, bufB, N, 128, 4, 5);

  // Layer 3: GAT(128->64, H=1, mean) ; output -> d_out
  run_gat_layer(bufB, W3, as3, ad3, bufA, bufC, sl, dl, menc, den,
                src, dst, N, E, 64, 1, 64, stream);
  gat_finalize_last<<<((N * 64) + 255) / 256, 256, 0, stream>>>(
      bufC, den, b3, (float*)d_out, N, 64);
}